// EmergentSpatialPropagation_9509057593769
// MI455X (gfx1250) — compile-verified
//
#include <hip/hip_runtime.h>
#include <stdint.h>

#define Wd 64
#define Hd 64
#define Dd 64
#define Cc 32
#define Bc 4
#define TY 8
#define PLW 68                 // 64 data + 1 left halo + 3 pad -> 272B row stride (16B aligned)
#define PLH 10                 // TY + 2 halo rows
#define PLANE (PLW * PLH)      // 680 floats
#define NBUF 4                 // 4 rotating z-planes (z halos live in zeroed registers)
#define STEPS 8

typedef unsigned int v4u __attribute__((ext_vector_type(4)));
typedef int          v8i __attribute__((ext_vector_type(8)));
typedef int          v4i __attribute__((ext_vector_type(4)));

#if defined(__has_builtin)
#if __has_builtin(__builtin_amdgcn_tensor_load_to_lds) && __has_builtin(__builtin_amdgcn_s_wait_tensorcnt)
#define USE_TDM 1
#endif
#endif

// Exact-erf GELU: 0.5*x*(1+erf(x/sqrt(2))), erf via Abramowitz-Stegun 7.1.26.
// rcp + exp2 run on the transcendental pipe and co-execute with the FMA stream.
__device__ __forceinline__ float gelu_exact(float x) {
  float u  = x * 0.70710678118654752f;
  float au = __builtin_fabsf(u);
  float t  = __builtin_amdgcn_rcpf(__builtin_fmaf(0.3275911f, au, 1.0f));
  float p  = __builtin_fmaf(t, 1.061405429f, -1.453152027f);
  p = __builtin_fmaf(t, p, 1.421413741f);
  p = __builtin_fmaf(t, p, -0.284496736f);
  p = __builtin_fmaf(t, p, 0.254829592f);
  p *= t;
  float e  = __builtin_amdgcn_exp2f(-au * au * 1.4426950408889634f);
  float er = __builtin_fmaf(-p, e, 1.0f);
  er = (u < 0.0f) ? -er : er;
  return 0.5f * x * (1.0f + er);
}

#ifdef USE_TDM
// One 2-D TDM tile: rows x 64 fp32 from global (row stride 64) into LDS with
// +4-dword padding after every 64 dwords (=> LDS row stride 68 dwords).
// Descriptor bitfields per CDNA5 ISA ch.8; 6-arg clang-23 builtin.
__device__ __forceinline__ void tdm_load_plane(const float* grow0, unsigned lds_byte, int rows) {
  uint64_t ga = (uint64_t)(uintptr_t)grow0;
  v4u g0;
  g0[0] = 1u;                                                  // count=1, is_restore=0, no gather
  g0[1] = lds_byte;                                            // lds_addr [63:32]
  g0[2] = (unsigned)(ga & 0xFFFFFFFFull);                      // global_addr lo
  g0[3] = (unsigned)((ga >> 32) & 0x01FFFFFFull) | (2u << 30); // addr[56:32] | type=2
  v8i g1;
  g1[0] = (2 << 16) | (1 << 20) | (5 << 22) | (3 << 25);       // 4B elems, pad_en, every 64dw, pad 4dw
  g1[1] = (int)(64u << 16);                                    // tensor_dim0 = 64
  g1[2] = (int)((unsigned)rows << 16);                         // tensor_dim1 = rows
  g1[3] = (int)(64u << 16);                                    // tile_dim0 = 64
  g1[4] = rows;                                                // tile_dim1 = rows; tile_dim2 = 0
  g1[5] = 64;                                                  // tensor_dim0_stride = 64 elems
  g1[6] = 0;
  g1[7] = 0;
  v4i zz4 = {0, 0, 0, 0};
  v8i zz8 = {0, 0, 0, 0, 0, 0, 0, 0};
  __builtin_amdgcn_tensor_load_to_lds(g0, g1, zz4, zz4, zz8, 0);
}
#endif

__device__ __forceinline__ void load_plane_sync(float* lds, int buf, const float* grow0,
                                                int rows, int l0, int tid) {
  // Fallback path: cooperative global->LDS copy of in-bounds cells only.
  const int n = rows * Wd;
  for (int i = tid; i < n; i += 128) {
    int r = i >> 6;
    int x = i & 63;
    lds[buf * PLANE + (l0 + r) * PLW + 1 + x] = grow0[r * Wd + x];
  }
}

// ---- register-window helpers (W = float[18]: 3 rows x 6 x-values) ----
#define LOADROW6(W, P, r) do {                                          \
    const float* _p = (P) + (ty + (r)) * PLW + xb;                      \
    float4 _q  = *(const float4*)_p;                                    \
    float2 _q2 = *(const float2*)(_p + 4);                              \
    W[(r)*6+0] = _q.x;  W[(r)*6+1] = _q.y;  W[(r)*6+2] = _q.z;          \
    W[(r)*6+3] = _q.w;  W[(r)*6+4] = _q2.x; W[(r)*6+5] = _q2.y;         \
  } while (0)

#define LOADWIN(W, P) do { LOADROW6(W, P, 0); LOADROW6(W, P, 1); LOADROW6(W, P, 2); } while (0)

#define ZEROWIN(W) do {                                                 \
    _Pragma("unroll") for (int _i = 0; _i < 18; ++_i) W[_i] = 0.0f;     \
  } while (0)

#define TAPS(W, r, wbase) do {                                                              \
    float _w0 = w[(wbase) + 0], _w1 = w[(wbase) + 1], _w2 = w[(wbase) + 2];                 \
    a0 = __builtin_fmaf(_w0, W[(r)*6+0], a0); a1 = __builtin_fmaf(_w0, W[(r)*6+1], a1);     \
    a2 = __builtin_fmaf(_w0, W[(r)*6+2], a2); a3 = __builtin_fmaf(_w0, W[(r)*6+3], a3);     \
    a0 = __builtin_fmaf(_w1, W[(r)*6+1], a0); a1 = __builtin_fmaf(_w1, W[(r)*6+2], a1);     \
    a2 = __builtin_fmaf(_w1, W[(r)*6+3], a2); a3 = __builtin_fmaf(_w1, W[(r)*6+4], a3);     \
    a0 = __builtin_fmaf(_w2, W[(r)*6+2], a0); a1 = __builtin_fmaf(_w2, W[(r)*6+3], a1);     \
    a2 = __builtin_fmaf(_w2, W[(r)*6+4], a2); a3 = __builtin_fmaf(_w2, W[(r)*6+5], a3);     \
  } while (0)

#define COMPUTE_STORE(PM, PC, PN, Z) do {                               \
    float a0 = PC[7], a1 = PC[8], a2 = PC[9], a3 = PC[10];  /* residual: x itself */ \
    TAPS(PM, 0, 0);  TAPS(PM, 1, 3);  TAPS(PM, 2, 6);   /* kz = 0 */    \
    TAPS(PC, 0, 9);  TAPS(PC, 1, 12); TAPS(PC, 2, 15);  /* kz = 1 */    \
    TAPS(PN, 0, 18); TAPS(PN, 1, 21); TAPS(PN, 2, 24);  /* kz = 2 */    \
    float4 _o;                                                          \
    _o.x = gelu_exact(a0); _o.y = gelu_exact(a1);                       \
    _o.z = gelu_exact(a2); _o.w = gelu_exact(a3);                       \
    *(float4*)(dst + chanBase + (size_t)(Z) * (Hd * Wd) + outRow + xb) = _o; \
  } while (0)

__global__ __launch_bounds__(128)
void esp_step_kernel(const float* __restrict__ src, const float* __restrict__ wgt,
                     float* __restrict__ dst) {
  __shared__ float lds[NBUF * PLANE];
  const int tx  = threadIdx.x;            // 0..15, 4 outputs each in x
  const int ty  = threadIdx.y;            // 0..7, one output row each
  const int tid = ty * 16 + tx;
  const int c   = blockIdx.y;
  const int b   = blockIdx.z;
  const int yb  = blockIdx.x * TY;
  const size_t chanBase = (size_t)(b * Cc + c) * (Dd * Hd * Wd);
  const size_t outRow   = (size_t)(yb + ty) * Wd;

  // Per-channel 3x3x3 weights: uniform across the block -> SGPRs via s_load.
  float w[27];
#pragma unroll
  for (int k = 0; k < 27; ++k) w[k] = wgt[c * 27 + k];

  // Zero all LDS buffers once: x/y halo cells are z-invariant OOB and stay 0;
  // the TDM only ever writes in-bounds cells afterwards.
  {
    float4 z4 = make_float4(0.f, 0.f, 0.f, 0.f);
    float4* l4 = (float4*)lds;
    for (int i = tid; i < (NBUF * PLANE) / 4; i += 128) l4[i] = z4;
  }
  __syncthreads();

  const int l0   = (yb == 0) ? 1 : 0;                 // first in-bounds halo row
  const int lend = (yb + TY == Hd) ? (PLH - 1) : PLH;
  const int rows = lend - l0;
  const int gy0  = yb + l0 - 1;                       // global y of first loaded row

#ifdef USE_TDM
  const int waveFirstTid = __builtin_amdgcn_readfirstlane(tid);  // scalar branch: wave 0 only
  if (waveFirstTid == 0) {
    tdm_load_plane(src + chanBase + (size_t)gy0 * Wd,
                   (unsigned)((0 * PLANE + l0 * PLW + 1) * 4), rows);
    tdm_load_plane(src + chanBase + (size_t)(Hd * Wd) + (size_t)gy0 * Wd,
                   (unsigned)((1 * PLANE + l0 * PLW + 1) * 4), rows);
  }
#else
  load_plane_sync(lds, 0, src + chanBase + (size_t)gy0 * Wd, rows, l0, tid);
  load_plane_sync(lds, 1, src + chanBase + (size_t)(Hd * Wd) + (size_t)gy0 * Wd, rows, l0, tid);
#endif

  const int xb = tx * 4;

  // FETCH(Z): make plane Z+1 LDS-resident; keep the DMA of plane Z+2 in flight.
#ifdef USE_TDM
#define FETCH(Z) do {                                                                  \
    __syncthreads();                                                                   \
    if (waveFirstTid == 0) {                                                           \
      if ((Z) <= Dd - 3) {                                                             \
        tdm_load_plane(src + chanBase + (size_t)((Z) + 2) * (Hd * Wd) + (size_t)gy0 * Wd, \
                       (unsigned)(((((Z) + 2) & 3) * PLANE + l0 * PLW + 1) * 4), rows);\
        __builtin_amdgcn_s_wait_tensorcnt(1);                                          \
      } else {                                                                         \
        __builtin_amdgcn_s_wait_tensorcnt(0);                                          \
      }                                                                                \
    }                                                                                  \
    __syncthreads();                                                                   \
  } while (0)
#else
#define FETCH(Z) do {                                                                  \
    __syncthreads();                                                                   \
    if ((Z) <= Dd - 3)                                                                 \
      load_plane_sync(lds, ((Z) + 2) & 3,                                              \
                      src + chanBase + (size_t)((Z) + 2) * (Hd * Wd) + (size_t)gy0 * Wd, \
                      rows, l0, tid);                                                  \
    __syncthreads();                                                                   \
  } while (0)
#endif

// One z-step with rotated register windows: refresh only PN (plane Z+1).
#define STEP(Z, PM, PC, PN) do {                                        \
    FETCH(Z);                                                           \
    if ((Z) + 1 < Dd) { LOADWIN(PN, lds + (((Z) + 1) & 3) * PLANE); }   \
    else              { ZEROWIN(PN); }                                  \
    COMPUTE_STORE(PM, PC, PN, Z);                                       \
  } while (0)

  float WA[18], WB[18], WC[18];

  // z = 0: WA = plane -1 (zeros), WB = plane 0, WC = plane 1.
  FETCH(0);                               // issues plane 2; guarantees planes 0,1 resident
  ZEROWIN(WA);
  LOADWIN(WB, lds + 0 * PLANE);
  LOADWIN(WC, lds + 1 * PLANE);
  COMPUTE_STORE(WA, WB, WC, 0);

  // z = 1..63 as 21 exact triples with static role rotation (period 3).
#pragma unroll 1
  for (int zb = 1; zb <= Dd - 3; zb += 3) {
    STEP(zb,     WB, WC, WA);
    STEP(zb + 1, WC, WA, WB);
    STEP(zb + 2, WA, WB, WC);   // zb+2 == 63 on the last triple -> PN zero-window
  }
}

extern "C" void kernel_launch(void* const* d_in, const int* in_sizes, int n_in,
                              void* d_out, int out_size, void* d_ws, size_t ws_size,
                              hipStream_t stream) {
  const float* x0  = (const float*)d_in[0];   // cube_states [4,32,64,64,64] f32
  const float* wgt = (const float*)d_in[1];   // weight [32,1,3,3,3] f32
  float* out = (float*)d_out;
  float* ws  = (float*)d_ws;                  // >= 128 MB ping buffer

  dim3 grid(Hd / TY, Cc, Bc);   // 8 x 32 x 4 = 1024 workgroups
  dim3 block(16, TY, 1);        // 128 threads = 4 waves (wave32)

  const float* srcp = x0;
  for (int s = 0; s < STEPS; ++s) {
    float* dstp = (s & 1) ? out : ws;         // even->ws, odd->out; step 8 ends in d_out
    esp_step_kernel<<<grid, block, 0, stream>>>(srcp, wgt, dstp);
    srcp = dstp;
  }
  (void)in_sizes; (void)n_in; (void)out_size; (void)ws_size;
}